// SphericalExpansion_23330262352080
// MI455X (gfx1250) — compile-verified
//
#include <hip/hip_runtime.h>
#include <hip/hip_bf16.h>

typedef __attribute__((ext_vector_type(2))) float v2f;
typedef __attribute__((ext_vector_type(8))) float v8f;

#define N_MAXB 8
#define FEAT 512
#define CUTOFF_F 5.0f

__device__ __forceinline__ void atomic_add_f32(float* p, float v) {
#if __has_builtin(__builtin_amdgcn_global_atomic_fadd_f32)
  typedef float __attribute__((address_space(1))) gfloat;
  __builtin_amdgcn_global_atomic_fadd_f32((gfloat*)p, v);
#else
  __hip_atomic_fetch_add(p, v, __ATOMIC_RELAXED, __HIP_MEMORY_SCOPE_AGENT);
#endif
}

// Atomic add at: uniform base + zext(32-bit voffset) + immediate.
// The +imm is done in the pointer (no-wrap) domain so it folds into the
// 24-bit immediate offset of the saddr-form global atomic.
__device__ __forceinline__ void atomic_add_su(float* __restrict__ base,
                                              unsigned voff, int immBytes,
                                              float v) {
  atomic_add_f32((float*)((char*)base + (size_t)voff + (size_t)immBytes), v);
}

// Per-edge math: returns output base offset (atom*512 + species*8),
// fills Y[16] (global m index) and rb[8].
__device__ __forceinline__ int edge_compute(
    const float* __restrict__ pos, const int* __restrict__ centers,
    const int* __restrict__ neighbors, const int* __restrict__ species,
    int e, float Y[16], float rb[8])
{
  const int c  = centers[e];
  const int nb = neighbors[e];
  const int sp = species[e];
  const float x = pos[3*nb+0] - pos[3*c+0];
  const float y = pos[3*nb+1] - pos[3*c+1];
  const float z = pos[3*nb+2] - pos[3*c+2];
  const float r2 = x*x + y*y + z*z + 1e-12f;
  const float r  = sqrtf(r2);
  const float ir = 1.0f / r;
  const float xn = x*ir, yn = y*ir, zn = z*ir;
  const float x2 = xn*xn, y2 = yn*yn, z2 = zn*zn;

  Y[0]  = 0.28209479177387814f;
  Y[1]  = 0.4886025119029199f * yn;
  Y[2]  = 0.4886025119029199f * zn;
  Y[3]  = 0.4886025119029199f * xn;
  Y[4]  = 1.0925484305920792f * xn * yn;
  Y[5]  = 1.0925484305920792f * yn * zn;
  Y[6]  = 0.31539156525252005f * (2.0f*z2 - x2 - y2);
  Y[7]  = 1.0925484305920792f * xn * zn;
  Y[8]  = 0.5462742152960396f * (x2 - y2);
  Y[9]  = 0.5900435899266435f * yn * (3.0f*x2 - y2);
  Y[10] = 2.890611442640554f  * xn * yn * zn;
  Y[11] = 0.4570457994644658f * yn * (4.0f*z2 - x2 - y2);
  Y[12] = 0.3731763325901154f * zn * (2.0f*z2 - 3.0f*x2 - 3.0f*y2);
  Y[13] = 0.4570457994644658f * xn * (4.0f*z2 - x2 - y2);
  Y[14] = 1.445305721320277f  * zn * (x2 - y2);
  Y[15] = 0.5900435899266435f * xn * (x2 - 3.0f*y2);

  const float theta = 3.14159265358979323846f * r * (1.0f / CUTOFF_F);
  float s1, c1;
  __sincosf(theta, &s1, &c1);
  const float fc    = (r < CUTOFF_F) ? (0.5f * (c1 + 1.0f)) : 0.0f;
  const float scale = 0.6324555320336759f * ir * fc;  // sqrt(2/rc)/r * fc
  const float twoc  = 2.0f * c1;
  float sk_1 = s1;
  float sk   = twoc * s1;
  rb[0] = scale * sk_1;
  rb[1] = scale * sk;
#pragma unroll
  for (int n = 2; n < 8; ++n) {
    const float sk1 = twoc * sk - sk_1;
    sk_1 = sk; sk = sk1;
    rb[n] = scale * sk;
  }
  return c * FEAT + sp * N_MAXB;   // out offset (floats) = base + 32*m + n
}

__global__ void zero_out_kernel(float4* __restrict__ out, int n4) {
  int i = blockIdx.x * blockDim.x + threadIdx.x;
  if (i < n4) out[i] = make_float4(0.f, 0.f, 0.f, 0.f);
}

// Main kernel: only FULL blocks of 256 edges -> no bounds/validity checks.
__global__ __launch_bounds__(256) void sph_expansion_kernel(
    const float* __restrict__ pos,
    const int*   __restrict__ centers,
    const int*   __restrict__ neighbors,
    const int*   __restrict__ species,
    float*       __restrict__ out)
{
  // Padded strides (17, 9 odd) -> conflict-free LDS banks.
  // Rbuf rows 256..287 are a zero region used as the B-operand "zero slot":
  // per-iteration offsets (2i rows) starting at row 256/257 stay inside it.
  __shared__ float    Ybuf[256][17];
  __shared__ float    Rbuf[288][9];
  __shared__ unsigned Abuf[256];    // pre-scaled BYTE offsets into out

  const int t = threadIdx.x;
  const int e = blockIdx.x * 256 + t;

  // Zero-fill the 32 pad rows of Rbuf (288 dwords).
  {
    float* pad = &Rbuf[256][0];
    for (int i = t; i < 32 * 9; i += 256) pad[i] = 0.0f;
  }

  float Y[16], rb[8];
  const int abase = edge_compute(pos, centers, neighbors, species, e, Y, rb);

#pragma unroll
  for (int i = 0; i < 16; ++i) Ybuf[t][i] = Y[i];
#pragma unroll
  for (int i = 0; i < 8; ++i)  Rbuf[t][i] = rb[i];
  Abuf[t] = (unsigned)abase * 4u;   // byte offset
  __syncthreads();

  // WMMA phase: 2 edges per V_WMMA_F32_16X16X4_F32.
  //  A: VGPR0/1 low lanes = Y_e0/Y_e1 (K0/K1). High lanes (K2/K3) load
  //     garbage -- harmless because B rows K2/K3 are zero.
  //  B: VGPR0 = row K0 = [rb_e0, 0...], VGPR1 = row K1 = [0x8, rb_e1, 0...];
  //     inactive lanes read the zeroed Rbuf pad via a pre-selected base.
  const int lane  = t & 31;
  const int wbase = (t >> 5) * 32;
  const int col   = lane & 15;       // N index
  const int n     = col & 7;         // radial index within an edge
  const int mhi   = (lane >> 4) * 8; // D rows v / v+8 across lane halves

  const float*    rA0 = &Ybuf[wbase + 0][col];
  const float*    rA1 = &Ybuf[wbase + 1][col];
  const float*    rB0 = (lane < 8)              ? &Rbuf[wbase + 0][lane]
                                                : &Rbuf[256][lane & 7];
  const float*    rB1 = ((unsigned)(lane - 8) < 8u) ? &Rbuf[wbase + 1][lane - 8]
                                                    : &Rbuf[257][lane & 7];
  const unsigned* rAb = &Abuf[wbase + ((col < 8) ? 0 : 1)];
  const unsigned laneByte = (unsigned)(n + 32 * mhi) * 4u;

#pragma unroll
  for (int i = 0; i < 16; ++i) {
    v2f a, b;
    a.x = rA0[i * 2 * 17];
    a.y = rA1[i * 2 * 17];
    b.x = rB0[i * 2 * 9];
    b.y = rB1[i * 2 * 9];
    v8f cacc = {};
    v8f d = __builtin_amdgcn_wmma_f32_16x16x4_f32(
        false, a, false, b, (short)0, cacc, false, false);

    const unsigned off = rAb[i * 2] + laneByte;   // u32 byte voffset
#pragma unroll
    for (int v = 0; v < 8; ++v)
      atomic_add_su(out, off, 128 * v, d[v]);     // imm folds: +32 floats/m
  }
}

// Tail kernel: remaining (<256) edges, plain VALU path.
__global__ __launch_bounds__(256) void sph_tail_kernel(
    const float* __restrict__ pos,
    const int*   __restrict__ centers,
    const int*   __restrict__ neighbors,
    const int*   __restrict__ species,
    float*       __restrict__ out,
    int start, int nEdges)
{
  const int e = start + (int)threadIdx.x;
  if (e >= nEdges) return;
  float Y[16], rb[8];
  const int ab = edge_compute(pos, centers, neighbors, species, e, Y, rb);
  const unsigned ab4 = (unsigned)ab * 4u;
#pragma unroll
  for (int m = 0; m < 16; ++m)
#pragma unroll
    for (int n = 0; n < 8; ++n)
      atomic_add_su(out, ab4, (32 * m + n) * 4, Y[m] * rb[n]);
}

extern "C" void kernel_launch(void* const* d_in, const int* in_sizes, int n_in,
                              void* d_out, int out_size, void* d_ws, size_t ws_size,
                              hipStream_t stream) {
  const float* pos       = (const float*)d_in[0];
  const int*   centers   = (const int*)d_in[1];
  const int*   neighbors = (const int*)d_in[2];
  const int*   species   = (const int*)d_in[3];
  float*       out       = (float*)d_out;
  const int nEdges = in_sizes[1];

  // out_size = N_ATOMS*512, divisible by 4.
  const int n4 = out_size / 4;
  zero_out_kernel<<<(n4 + 255) / 256, 256, 0, stream>>>((float4*)out, n4);

  const int nFull = nEdges / 256;
  const int rem   = nEdges - nFull * 256;
  if (nFull > 0)
    sph_expansion_kernel<<<nFull, 256, 0, stream>>>(pos, centers, neighbors,
                                                    species, out);
  if (rem > 0)
    sph_tail_kernel<<<1, 256, 0, stream>>>(pos, centers, neighbors, species,
                                           out, nFull * 256, nEdges);
}